// zoomer_89111981457716
// MI455X (gfx1250) — compile-verified
//
#include <hip/hip_runtime.h>
#include <hip/hip_bf16.h>

#define U_NUM 7000
#define M_NUM 3000
#define N_NUM 10000
#define B_NUM 4096
#define MAXD  128
#define LRELU 0.2f

typedef float v2f __attribute__((ext_vector_type(2)));
typedef float v4f __attribute__((ext_vector_type(4)));
typedef float v8f __attribute__((ext_vector_type(8)));

// ---------------------------------------------------------------------------
// Pack B matrices for WMMA: Bp1[32x16] = [W_h0 | W_h1], Bp2[16x16] = W_out
// ---------------------------------------------------------------------------
__global__ void pack_b(const float* __restrict__ Wh0, const float* __restrict__ Wh1,
                       const float* __restrict__ Wout,
                       float* __restrict__ Bp1, float* __restrict__ Bp2) {
  int t = threadIdx.x;                       // 512 threads
  if (t < 512) {
    int k = t >> 4, c = t & 15;
    Bp1[t] = (c < 8) ? Wh0[k * 8 + c] : Wh1[k * 8 + (c - 8)];
  }
  if (t < 256) Bp2[t] = Wout[t];
}

// ---------------------------------------------------------------------------
// Node features: x[N,32] = mapped user/movie embeddings (one wave per row)
// ---------------------------------------------------------------------------
__global__ void build_x(const int* __restrict__ gender, const int* __restrict__ age,
                        const int* __restrict__ job, const int* __restrict__ mtype,
                        const float* __restrict__ E_uid, const float* __restrict__ E_g,
                        const float* __restrict__ E_age, const float* __restrict__ E_job,
                        const float* __restrict__ E_mid, const float* __restrict__ E_mt,
                        const float* __restrict__ Wu, const float* __restrict__ bu,
                        const float* __restrict__ Wm, const float* __restrict__ bm,
                        float* __restrict__ x) {
  int lane = threadIdx.x & 31;
  int row  = blockIdx.x * 8 + (threadIdx.x >> 5);   // grid covers N exactly
  float acc;
  if (row < U_NUM) {
    int u = row;
    acc = bu[lane];
    #pragma unroll 4
    for (int k = 0; k < 32; ++k) acc += E_uid[u * 32 + k] * Wu[k * 32 + lane];
    int g = gender[u];
    #pragma unroll 4
    for (int k = 0; k < 16; ++k) acc += E_g[g * 16 + k] * Wu[(32 + k) * 32 + lane];
    int a = age[u];
    #pragma unroll 4
    for (int k = 0; k < 16; ++k) acc += E_age[a * 16 + k] * Wu[(48 + k) * 32 + lane];
    int j = job[u];
    #pragma unroll 4
    for (int k = 0; k < 16; ++k) acc += E_job[j * 16 + k] * Wu[(64 + k) * 32 + lane];
  } else {
    int m = row - U_NUM;
    acc = bm[lane];
    #pragma unroll 4
    for (int k = 0; k < 32; ++k) acc += E_mid[m * 32 + k] * Wm[k * 32 + lane];
    int g0 = mtype[m * 3 + 0], g1 = mtype[m * 3 + 1], g2 = mtype[m * 3 + 2];
    #pragma unroll 4
    for (int k = 0; k < 32; ++k) {
      float t = E_mt[g0 * 32 + k] + E_mt[g1 * 32 + k] + E_mt[g2 * 32 + k];
      acc += t * Wm[(32 + k) * 32 + lane];
    }
  }
  x[row * 32 + lane] = acc;
}

// ---------------------------------------------------------------------------
// Sparsify adjacency: one block per row, LDS-atomic compaction into ELL.
// adj is read exactly once (400 MB -> ~17 us at 23.3 TB/s; the dominant cost).
// 400 MB > 192 MB global L2, so stream it with TH=NT (non-temporal) loads to
// avoid evicting the ~6 MB hot working set (Wh/cols/deg) that the gather-
// heavy aggregation kernels need resident in L2.
// ---------------------------------------------------------------------------
__global__ void build_ell(const float* __restrict__ adj, int* __restrict__ cols,
                          int* __restrict__ deg) {
  __shared__ int cnt;
  __shared__ int list[MAXD];
  int row = blockIdx.x;
  if (threadIdx.x == 0) cnt = 0;
  __syncthreads();
  const v4f* arow = (const v4f*)(adj + (size_t)row * N_NUM);
  for (int c4 = threadIdx.x; c4 < N_NUM / 4; c4 += blockDim.x) {
    v4f v = __builtin_nontemporal_load(&arow[c4]);   // global_load_b128 TH_NT
    int base = c4 * 4;
    if (v.x > 0.f) { int p = atomicAdd(&cnt, 1); if (p < MAXD) list[p] = base + 0; }
    if (v.y > 0.f) { int p = atomicAdd(&cnt, 1); if (p < MAXD) list[p] = base + 1; }
    if (v.z > 0.f) { int p = atomicAdd(&cnt, 1); if (p < MAXD) list[p] = base + 2; }
    if (v.w > 0.f) { int p = atomicAdd(&cnt, 1); if (p < MAXD) list[p] = base + 3; }
  }
  __syncthreads();
  int d = cnt < MAXD ? cnt : MAXD;
  if (threadIdx.x == 0) deg[row] = d;
  for (int t = threadIdx.x; t < MAXD; t += blockDim.x)
    cols[row * MAXD + t] = (t < d) ? list[t] : 0;
}

// ---------------------------------------------------------------------------
// WMMA feature transform: Out[N,16] = In[N,KDIM] @ Bp[KDIM,16]
// One wave per 16-row tile; v_wmma_f32_16x16x4_f32 accumulation over K.
// A layout (ISA §7.12.2): lanes0-15 M=lane, K={k,k+1}; lanes16-31 K={k+2,k+3}.
// B/C row-striped across lanes (16x16 f32 layout).
// ---------------------------------------------------------------------------
template <int KDIM>
__global__ void wmma_transform(const float* __restrict__ In, const float* __restrict__ Bp,
                               float* __restrict__ Out) {
  int lane = threadIdx.x;               // 32 threads, EXEC all ones, no divergence
  int tile = blockIdx.x;
  int col  = lane & 15;
  int ko   = (lane < 16) ? 0 : 2;
  int row  = tile * 16 + col;
  v8f c = {};
  #pragma unroll
  for (int k = 0; k < KDIM; k += 4) {
    v2f a, b;
    a.x = In[row * KDIM + k + ko];
    a.y = In[row * KDIM + k + ko + 1];
    b.x = Bp[(k + ko) * 16 + col];
    b.y = Bp[(k + ko + 1) * 16 + col];
    c = __builtin_amdgcn_wmma_f32_16x16x4_f32(false, a, false, b, (short)0, c,
                                              false, false);
  }
  #pragma unroll
  for (int r = 0; r < 8; ++r) {
    int m = tile * 16 + r + ((lane < 16) ? 0 : 8);
    Out[m * 16 + col] = c[r];
  }
}

// ---------------------------------------------------------------------------
// GAT attention aggregation over ELL neighbors: one wave32 per row.
// Exact two-pass softmax (max via shfl_xor butterfly); fused LeakyReLU + ELU.
// Wh row stride is always 16 (2 heads x 8 or 1 head x 16).
// ---------------------------------------------------------------------------
template <int HDIM, int NHEADS>
__global__ void gat_aggregate(const float* __restrict__ Wh, const int* __restrict__ cols,
                              const int* __restrict__ deg,
                              const float* __restrict__ a0, const float* __restrict__ a1,
                              float* __restrict__ outbuf) {
  const int SLOTS = MAXD / 32;
  int lane = threadIdx.x & 31;
  int row  = blockIdx.x * 8 + (threadIdx.x >> 5);   // grid covers N exactly

  float asrc[NHEADS][HDIM], adst[NHEADS][HDIM];
  #pragma unroll
  for (int h = 0; h < NHEADS; ++h) {
    const float* av = (h == 0) ? a0 : a1;
    #pragma unroll
    for (int c = 0; c < HDIM; ++c) { asrc[h][c] = av[c]; adst[h][c] = av[HDIM + c]; }
  }

  float fsrc[NHEADS];
  #pragma unroll
  for (int h = 0; h < NHEADS; ++h) {
    float s = 0.f;
    #pragma unroll
    for (int c = 0; c < HDIM; ++c) s += Wh[row * 16 + h * HDIM + c] * asrc[h][c];
    fsrc[h] = s;
  }

  int d = deg[row];
  const int* cl = cols + row * MAXD;
  int   nj[SLOTS];
  float e[NHEADS][SLOTS];
  float mx[NHEADS];
  #pragma unroll
  for (int h = 0; h < NHEADS; ++h) mx[h] = -3.0e38f;

  #pragma unroll
  for (int s = 0; s < SLOTS; ++s) {
    int idx = lane + s * 32;
    nj[s] = -1;
    #pragma unroll
    for (int h = 0; h < NHEADS; ++h) e[h][s] = -3.0e38f;
    if (idx < d) {
      int j = cl[idx];
      nj[s] = j;
      #pragma unroll
      for (int h = 0; h < NHEADS; ++h) {
        float fd = 0.f;
        #pragma unroll
        for (int c = 0; c < HDIM; ++c) fd += Wh[j * 16 + h * HDIM + c] * adst[h][c];
        float ee = fsrc[h] + fd;
        ee = (ee > 0.f) ? ee : LRELU * ee;     // LeakyReLU
        e[h][s] = ee;
        mx[h] = fmaxf(mx[h], ee);
      }
    }
  }

  #pragma unroll
  for (int off = 16; off >= 1; off >>= 1)
    #pragma unroll
    for (int h = 0; h < NHEADS; ++h)
      mx[h] = fmaxf(mx[h], __shfl_xor(mx[h], off));

  float acc[NHEADS][HDIM], ssum[NHEADS];
  #pragma unroll
  for (int h = 0; h < NHEADS; ++h) {
    ssum[h] = 0.f;
    #pragma unroll
    for (int c = 0; c < HDIM; ++c) acc[h][c] = 0.f;
  }

  #pragma unroll
  for (int s = 0; s < SLOTS; ++s) {
    if (nj[s] >= 0) {
      #pragma unroll
      for (int h = 0; h < NHEADS; ++h) {
        float w = __expf(e[h][s] - mx[h]);
        ssum[h] += w;
        #pragma unroll
        for (int c = 0; c < HDIM; ++c)
          acc[h][c] += w * Wh[nj[s] * 16 + h * HDIM + c];
      }
    }
  }

  #pragma unroll
  for (int off = 16; off >= 1; off >>= 1) {
    #pragma unroll
    for (int h = 0; h < NHEADS; ++h) {
      ssum[h] += __shfl_xor(ssum[h], off);
      #pragma unroll
      for (int c = 0; c < HDIM; ++c) acc[h][c] += __shfl_xor(acc[h][c], off);
    }
  }

  if (lane < NHEADS * HDIM) {
    float v = 0.f, sden = 1.f;
    #pragma unroll
    for (int h = 0; h < NHEADS; ++h)
      #pragma unroll
      for (int c = 0; c < HDIM; ++c)
        if (lane == h * HDIM + c) { v = acc[h][c]; sden = ssum[h]; }
    v = v / sden;
    v = (v > 0.f) ? v : (__expf(v) - 1.f);     // ELU (hidden concat & outer both apply it)
    outbuf[row * 16 + lane] = v;
  }
}

// ---------------------------------------------------------------------------
// Final prediction: sigmoid(concat(gcn[uid], gcn[mid+U]) @ W_pred + b) * 5
// ---------------------------------------------------------------------------
__global__ void predict(const float* __restrict__ gcn, const int* __restrict__ uid,
                        const int* __restrict__ mid, const float* __restrict__ Wp,
                        const float* __restrict__ bp, float* __restrict__ out) {
  int b = blockIdx.x * blockDim.x + threadIdx.x;
  if (b >= B_NUM) return;
  const float* ue = gcn + (size_t)uid[b] * 16;
  const float* me = gcn + (size_t)(mid[b] + U_NUM) * 16;
  float s = bp[0];
  #pragma unroll
  for (int c = 0; c < 16; ++c) s += ue[c] * Wp[c];
  #pragma unroll
  for (int c = 0; c < 16; ++c) s += me[c] * Wp[16 + c];
  float p = 5.f / (1.f + __expf(-s));
  out[b]          = p;          // pred
  out[B_NUM + b]  = ceilf(p);   // pred_rating
}

// ---------------------------------------------------------------------------
extern "C" void kernel_launch(void* const* d_in, const int* in_sizes, int n_in,
                              void* d_out, int out_size, void* d_ws, size_t ws_size,
                              hipStream_t stream) {
  const int*   uid    = (const int*)d_in[0];
  const int*   mid    = (const int*)d_in[1];
  const int*   gender = (const int*)d_in[2];
  const int*   age    = (const int*)d_in[3];
  const int*   job    = (const int*)d_in[4];
  const int*   mtype  = (const int*)d_in[5];
  const float* adj    = (const float*)d_in[6];
  const float* E_uid  = (const float*)d_in[7];
  const float* E_g    = (const float*)d_in[8];
  const float* E_age  = (const float*)d_in[9];
  const float* E_job  = (const float*)d_in[10];
  const float* E_mid  = (const float*)d_in[11];
  const float* E_mt   = (const float*)d_in[12];
  const float* Wu     = (const float*)d_in[13];
  const float* bu     = (const float*)d_in[14];
  const float* Wm     = (const float*)d_in[15];
  const float* bm     = (const float*)d_in[16];
  const float* W_h0   = (const float*)d_in[17];
  const float* a_h0   = (const float*)d_in[18];
  const float* W_h1   = (const float*)d_in[19];
  const float* a_h1   = (const float*)d_in[20];
  const float* W_out  = (const float*)d_in[21];
  const float* a_out  = (const float*)d_in[22];
  const float* W_pred = (const float*)d_in[23];
  const float* b_pred = (const float*)d_in[24];

  float* ws   = (float*)d_ws;
  float* x    = ws;                  // N*32   = 320000 floats
  float* Wh   = ws + 320000;         // N*16
  float* hcat = ws + 480000;         // N*16
  float* Wh2  = ws + 640000;         // N*16
  float* gcn  = ws + 800000;         // N*16
  float* Bp1  = ws + 960000;         // 512
  float* Bp2  = ws + 960512;         // 256
  int*   cols = (int*)(ws + 960768); // N*128 ints (16B-aligned offset)
  int*   deg  = cols + (size_t)N_NUM * MAXD;
  float* out  = (float*)d_out;

  pack_b<<<1, 512, 0, stream>>>(W_h0, W_h1, W_out, Bp1, Bp2);
  build_x<<<N_NUM / 8, 256, 0, stream>>>(gender, age, job, mtype, E_uid, E_g, E_age,
                                         E_job, E_mid, E_mt, Wu, bu, Wm, bm, x);
  build_ell<<<N_NUM, 256, 0, stream>>>(adj, cols, deg);
  wmma_transform<32><<<N_NUM / 16, 32, 0, stream>>>(x, Bp1, Wh);
  gat_aggregate<8, 2><<<N_NUM / 8, 256, 0, stream>>>(Wh, cols, deg, a_h0, a_h1, hcat);
  wmma_transform<16><<<N_NUM / 16, 32, 0, stream>>>(hcat, Bp2, Wh2);
  gat_aggregate<16, 1><<<N_NUM / 8, 256, 0, stream>>>(Wh2, cols, deg, a_out, a_out, gcn);
  predict<<<B_NUM / 256, 256, 0, stream>>>(gcn, uid, mid, W_pred, b_pred, out);
}